// SDDH_2594160246860
// MI455X (gfx1250) — compile-verified
//
#include <hip/hip_runtime.h>
#include <hip/hip_bf16.h>
#include <math.h>

#define Bc    4
#define Cc    128
#define Hc    256
#define Wc    256
#define Nc    4000
#define NPOSc 8
#define KSc   3
#define CHc   16
#define KPATCH (Cc * KSc * KSc)   // 1152

typedef float v2f __attribute__((ext_vector_type(2)));
typedef float v8f __attribute__((ext_vector_type(8)));

// ---------------------------------------------------------------------------
// Kernel A: patch gather -> conv-as-GEMM (N x 1152 x 16) -> offsets.
// Gather-latency bound (random 3x3x128 patches, L2-resident); scalar VALU.
// 8 keypoints per 128-thread block.
// ---------------------------------------------------------------------------
__global__ void __launch_bounds__(128) sddh_offsets_kernel(
    const float* __restrict__ x, const float* __restrict__ kpts,
    const float* __restrict__ w1, const float* __restrict__ b1,
    const float* __restrict__ w2, const float* __restrict__ b2,
    float* __restrict__ out_off)
{
    __shared__ float patch[8][KPATCH];
    __shared__ float h1s[8][CHc];
    __shared__ int   cxs[8], cys[8];

    const int blocks_per_img = Nc / 8;            // 500
    const int b  = blockIdx.x / blocks_per_img;
    const int n0 = (blockIdx.x % blocks_per_img) * 8;
    const int t  = threadIdx.x;
    const float* xb = x + (size_t)b * Cc * Hc * Wc;

    if (t < 8) {
        const int n = n0 + t;
        const float kx = kpts[((size_t)b * Nc + n) * 2 + 0];
        const float ky = kpts[((size_t)b * Nc + n) * 2 + 1];
        const float wx = (kx * 0.5f + 0.5f) * (Wc - 1.0f);
        const float wy = (ky * 0.5f + 0.5f) * (Hc - 1.0f);
        const int rcx = (int)wx;                  // trunc (non-negative)
        const int rcy = (int)wy;
        int cx = (int)truncf((float)rcx - 0.5f);  // trunc(rc - KS/2 + 1)
        int cy = (int)truncf((float)rcy - 0.5f);
        cx = min(max(cx, 0), Wc - 1 - KSc);
        cy = min(max(cy, 0), Hc - 1 - KSc);
        cxs[t] = cx; cys[t] = cy;
    }
    __syncthreads();

    // cooperative patch gather: 8 * 1152 elements
    for (int i = t; i < 8 * KPATCH; i += 128) {
        const int kp = i / KPATCH;
        const int r  = i - kp * KPATCH;
        const int c  = r / (KSc * KSc);
        const int rr = r - c * (KSc * KSc);
        const int dy = rr / KSc;
        const int dx = rr - dy * KSc;
        patch[kp][r] = xb[(size_t)c * Hc * Wc + (size_t)(cys[kp] + dy) * Wc + (cxs[kp] + dx)];
    }
    __syncthreads();

    // h1 = relu(patch . w1 + b1): thread = (keypoint, out-channel)
    {
        const int kp = t >> 4;
        const int o  = t & 15;
        float s = b1[o];
        const float* wrow = w1 + (size_t)o * KPATCH;
        for (int r = 0; r < KPATCH; ++r) s = fmaf(patch[kp][r], wrow[r], s);
        h1s[kp][o] = fmaxf(s, 0.0f);
    }
    __syncthreads();

    // off = clip(h1 @ w2.T + b2, +-64); reshape (-1,2,8) -> (n,p,2)
    {
        const int kp = t >> 4;
        const int j  = t & 15;
        float s = b2[j];
        for (int o = 0; o < CHc; ++o) s = fmaf(h1s[kp][o], w2[j * CHc + o], s);
        const float mo = 64.0f;                   // max(H,W)/4
        s = fminf(fmaxf(s, -mo), mo);
        const int d = j >> 3;                     // coord (0=x,1=y)
        const int p = j & 7;                      // position
        const int n = n0 + kp;
        out_off[(((size_t)b * Nc + n) * NPOSc + p) * 2 + d] = s;
    }
}

// ---------------------------------------------------------------------------
// Kernel B: bilinear sample -> selu(Wsf@F) -> sum_p G^T@Agg[p] -> normalize.
// 16 keypoints per 128-thread block (4 waves). All GEMM work on
// v_wmma_f32_16x16x4_f32 (fp32 in == reference precision).
// Dynamic LDS: F[128][128] (64KB) + Gp[128][16] (8KB, reused as D stage) + 16.
// ---------------------------------------------------------------------------
__global__ void __launch_bounds__(128) sddh_desc_kernel(
    const float* __restrict__ x, const float* __restrict__ kpts,
    const float* __restrict__ wsf, const float* __restrict__ agg,
    const float* __restrict__ out_off, float* __restrict__ out_desc)
{
    extern __shared__ float lds[];
    float* F   = lds;                  // 128 x 128, F[c][col], col = kp*8+p
    float* Gp  = lds + 128 * 128;      // 128 x 16 (reused as D stage 16 x 128)
    float* nrm = Gp + 128 * 16;        // 16

    const int tiles_per_img = Nc / 16; // 250
    const int b    = blockIdx.x / tiles_per_img;
    const int n0   = (blockIdx.x % tiles_per_img) * 16;
    const int t    = threadIdx.x;
    const int wave = t >> 5;
    const int lane = t & 31;
    const int half = lane >> 4;
    const int l16  = lane & 15;
    const int mt0  = wave * 2;         // this wave's two 16-wide tiles
    const int mt1  = mt0 + 1;
    const float* xb = x + (size_t)b * Cc * Hc * Wc;

    // ---- Phase 1: bilinear sampling; thread t owns column col = t ----
    {
        const int kp = t >> 3;
        const int p  = t & 7;
        const int n  = n0 + kp;
        const float kx = kpts[((size_t)b * Nc + n) * 2 + 0];
        const float ky = kpts[((size_t)b * Nc + n) * 2 + 1];
        const float px = (kx * 0.5f + 0.5f) * (Wc - 1.0f)
                       + out_off[(((size_t)b * Nc + n) * NPOSc + p) * 2 + 0];
        const float py = (ky * 0.5f + 0.5f) * (Hc - 1.0f)
                       + out_off[(((size_t)b * Nc + n) * NPOSc + p) * 2 + 1];
        const float x0f = floorf(px), y0f = floorf(py);
        const int   x0 = (int)x0f,  y0 = (int)y0f;
        const float wx1 = px - x0f, wx0 = 1.0f - wx1;
        const float wy1 = py - y0f, wy0 = 1.0f - wy1;
        const int   xs[2]  = { x0, x0 + 1 };
        const int   ys[2]  = { y0, y0 + 1 };
        const float wxs[2] = { wx0, wx1 };
        const float wys[2] = { wy0, wy1 };
        float wgt[4]; int idx[4];
#pragma unroll
        for (int ty = 0; ty < 2; ++ty)
#pragma unroll
            for (int tx = 0; tx < 2; ++tx) {
                const int xi = xs[tx], yi = ys[ty];
                const bool valid = (xi >= 0) && (xi < Wc) && (yi >= 0) && (yi < Hc);
                const int xc = min(max(xi, 0), Wc - 1);
                const int yc = min(max(yi, 0), Hc - 1);
                wgt[ty * 2 + tx] = valid ? wxs[tx] * wys[ty] : 0.0f;
                idx[ty * 2 + tx] = yc * Wc + xc;
            }
        for (int c = 0; c < Cc; ++c) {
            const float* xcp = xb + (size_t)c * Hc * Wc;
            F[c * 128 + t] = xcp[idx[0]] * wgt[0] + xcp[idx[1]] * wgt[1]
                           + xcp[idx[2]] * wgt[2] + xcp[idx[3]] * wgt[3];
        }
    }
    __syncthreads();

    const v8f zero8 = {0.f, 0.f, 0.f, 0.f, 0.f, 0.f, 0.f, 0.f};
    v8f dacc0 = zero8, dacc1 = zero8;   // persistent D accumulators (16 x 32 cols)

    const float selu_scale = 1.0507009873554805f;
    const float selu_alpha = 1.6732632423543772f;

    for (int p = 0; p < NPOSc; ++p) {
        // ---- step 6: G_p = selu(Wsf @ F_p); wave owns row-tiles mt0, mt1 ----
        v8f acc0 = zero8, acc1 = zero8;
        for (int k0 = 0; k0 < Cc; k0 += 4) {
            const int klo = k0 + 2 * half;
            v2f bf, a0, a1;
            bf.x = F[(klo + 0) * 128 + l16 * 8 + p];   // B[k][n] = F_p[c][kp]
            bf.y = F[(klo + 1) * 128 + l16 * 8 + p];
            a0.x = wsf[(size_t)(mt0 * 16 + l16) * Cc + klo + 0];
            a0.y = wsf[(size_t)(mt0 * 16 + l16) * Cc + klo + 1];
            a1.x = wsf[(size_t)(mt1 * 16 + l16) * Cc + klo + 0];
            a1.y = wsf[(size_t)(mt1 * 16 + l16) * Cc + klo + 1];
            acc0 = __builtin_amdgcn_wmma_f32_16x16x4_f32(false, a0, false, bf,
                                                         (short)0, acc0, false, false);
            acc1 = __builtin_amdgcn_wmma_f32_16x16x4_f32(false, a1, false, bf,
                                                         (short)0, acc1, false, false);
        }
        __syncthreads();   // prior step-7 reads of Gp are done before overwrite
#pragma unroll
        for (int v = 0; v < 8; ++v) {
            float g0 = acc0[v], g1 = acc1[v];
            g0 = (g0 > 0.f) ? selu_scale * g0
                            : selu_scale * selu_alpha * (__expf(g0) - 1.0f);
            g1 = (g1 > 0.f) ? selu_scale * g1
                            : selu_scale * selu_alpha * (__expf(g1) - 1.0f);
            Gp[(mt0 * 16 + v + 8 * half) * 16 + l16] = g0;
            Gp[(mt1 * 16 + v + 8 * half) * 16 + l16] = g1;
        }
        __syncthreads();

        // ---- step 7: D += G_p^T @ Agg[p]; wave owns col-tiles mt0, mt1 ----
        const float* aggp = agg + (size_t)p * Cc * Cc;
        for (int k0 = 0; k0 < Cc; k0 += 4) {
            const int klo = k0 + 2 * half;
            v2f af, b0, b1;
            af.x = Gp[(klo + 0) * 16 + l16];           // A[m][k] = G_p[c][kp]
            af.y = Gp[(klo + 1) * 16 + l16];
            b0.x = aggp[(size_t)(klo + 0) * Cc + mt0 * 16 + l16];
            b0.y = aggp[(size_t)(klo + 1) * Cc + mt0 * 16 + l16];
            b1.x = aggp[(size_t)(klo + 0) * Cc + mt1 * 16 + l16];
            b1.y = aggp[(size_t)(klo + 1) * Cc + mt1 * 16 + l16];
            dacc0 = __builtin_amdgcn_wmma_f32_16x16x4_f32(false, af, false, b0,
                                                          (short)0, dacc0, false, false);
            dacc1 = __builtin_amdgcn_wmma_f32_16x16x4_f32(false, af, false, b1,
                                                          (short)0, dacc1, false, false);
        }
    }
    __syncthreads();       // last step-7 reads of Gp done; reuse as D stage

    // ---- stage D (16 x 128) into LDS, then normalize rows ----
    float* Dst = Gp;
#pragma unroll
    for (int v = 0; v < 8; ++v) {
        const int m = v + 8 * half;    // keypoint row within tile
        Dst[m * 128 + mt0 * 16 + l16] = dacc0[v];
        Dst[m * 128 + mt1 * 16 + l16] = dacc1[v];
    }
    __syncthreads();
    if (t < 16) {
        float s = 0.0f;
        for (int d = 0; d < Cc; ++d) { const float v = Dst[t * 128 + d]; s += v * v; }
        nrm[t] = fmaxf(sqrtf(s), 1e-12f);
    }
    __syncthreads();
    {
        const int row = t >> 3;        // 8 threads per keypoint row
        const int j   = t & 7;
        const int n   = n0 + row;
        const float inv = 1.0f / nrm[row];
        float* outp = out_desc + ((size_t)b * Nc + n) * Cc;
        for (int d = j * 16; d < j * 16 + 16; ++d)
            outp[d] = Dst[row * 128 + d] * inv;
    }
}

// ---------------------------------------------------------------------------
extern "C" void kernel_launch(void* const* d_in, const int* in_sizes, int n_in,
                              void* d_out, int out_size, void* d_ws, size_t ws_size,
                              hipStream_t stream) {
    const float* x    = (const float*)d_in[0];
    const float* kpts = (const float*)d_in[1];
    const float* w1   = (const float*)d_in[2];
    const float* b1   = (const float*)d_in[3];
    const float* w2   = (const float*)d_in[4];
    const float* b2   = (const float*)d_in[5];
    const float* wsf  = (const float*)d_in[6];
    const float* agg  = (const float*)d_in[7];

    float* out_desc = (float*)d_out;                       // (B, N, C)
    float* out_off  = out_desc + (size_t)Bc * Nc * Cc;     // (B, N, NPOS, 2)

    sddh_offsets_kernel<<<Bc * (Nc / 8), 128, 0, stream>>>(
        x, kpts, w1, b1, w2, b2, out_off);

    const size_t shmem = (size_t)(128 * 128 + 128 * 16 + 16) * sizeof(float); // ~72KB
    (void)hipFuncSetAttribute((const void*)sddh_desc_kernel,
                              hipFuncAttributeMaxDynamicSharedMemorySize,
                              (int)shmem);
    sddh_desc_kernel<<<Bc * (Nc / 16), 128, shmem, stream>>>(
        x, kpts, wsf, agg, out_off, out_desc);
}